// QuantumGenerator_67370857005701
// MI455X (gfx1250) — compile-verified
//
#include <hip/hip_runtime.h>
#include <hip/hip_bf16.h>

#define N_LAYERS 2
#define N_QUBITS 15
#define DIM 32768            // 2^15
#define SIM_THREADS 1024     // 32 waves on one WGP

typedef float v2f __attribute__((ext_vector_type(2)));
typedef float v8f __attribute__((ext_vector_type(8)));

__device__ __forceinline__ float2 cmul(float2 a, float2 b) {
    return make_float2(a.x * b.x - a.y * b.y, a.x * b.y + a.y * b.x);
}

// u0*a + u1*b  (complex)
__device__ __forceinline__ float2 cfma2(float2 u0, float2 a, float2 u1, float2 b) {
    return make_float2(u0.x * a.x - u0.y * a.y + u1.x * b.x - u1.y * b.y,
                       u0.x * a.y + u0.y * a.x + u1.x * b.y + u1.y * b.x);
}

// ---------------------------------------------------------------------------
// Kernel 1: full 15-qubit state-vector simulation inside one WGP's LDS.
//   - state: 32768 x complex64 = 256 KB LDS (<= 320 KB/WGP on CDNA5)
//   - RZ*RY*RX fused to one 2x2 complex gate per qubit (30 sweeps total)
//   - all adjacent CZs of a layer fused to one sign sweep:
//       sign(idx) = (-1)^popcount(idx & (idx>>1))
//   - <Z_q> reduced wave32-local via __shfl_down, then ds_add_f32 atomics
// Writes 15 scaled expvals to ev_out.
// ---------------------------------------------------------------------------
extern __shared__ float smem_dyn[];

__global__ void __launch_bounds__(SIM_THREADS)
qsim_kernel(const float* __restrict__ weights,   // [2,15,3]
            const float* __restrict__ scale,     // [1]
            float* __restrict__ ev_out)          // [15]
{
    float2* st  = reinterpret_cast<float2*>(smem_dyn);  // DIM complex amplitudes
    float*  acc = smem_dyn + 2 * DIM;                   // 16-float accumulator

    const int tid = threadIdx.x;

    // |00...0>
    for (int i = tid; i < DIM; i += SIM_THREADS)
        st[i] = make_float2(i == 0 ? 1.0f : 0.0f, 0.0f);
    if (tid < 16) acc[tid] = 0.0f;
    __syncthreads();

    for (int l = 0; l < N_LAYERS; ++l) {
        for (int w = 0; w < N_QUBITS; ++w) {
            // Build U = Rz(t2) * Ry(t1) * Rx(t0), half angles.
            const float* th = weights + (l * N_QUBITS + w) * 3;
            float s0, c0, s1, c1, s2, c2;
            __sincosf(th[0] * 0.5f, &s0, &c0);
            __sincosf(th[1] * 0.5f, &s1, &c1);
            __sincosf(th[2] * 0.5f, &s2, &c2);
            // m = Ry*Rx (Ry real):
            float2 m00 = make_float2( c1 * c0,  s1 * s0);
            float2 m01 = make_float2(-s1 * c0, -c1 * s0);
            float2 m10 = make_float2( s1 * c0, -c1 * s0);
            float2 m11 = make_float2( c1 * c0, -s1 * s0);
            // U = diag(e^{-i t2/2}, e^{+i t2/2}) * m
            float2 e0 = make_float2(c2, -s2);
            float2 e1 = make_float2(c2,  s2);
            float2 u00 = cmul(e0, m00), u01 = cmul(e0, m01);
            float2 u10 = cmul(e1, m10), u11 = cmul(e1, m11);

            const int shift = 14 - w;           // wire w lives at bit (14-w)
            const int rest  = 1 << shift;
            for (int p = tid; p < DIM / 2; p += SIM_THREADS) {
                int i    = p >> shift;
                int j    = p & (rest - 1);
                int idx0 = (i << (shift + 1)) | j;
                int idx1 = idx0 | rest;
                float2 a0 = st[idx0];
                float2 a1 = st[idx1];
                st[idx0] = cfma2(u00, a0, u01, a1);
                st[idx1] = cfma2(u10, a0, u11, a1);
            }
            __syncthreads();
        }
        // Fused CZ layer: flip sign where an odd number of adjacent bit pairs are 11.
        for (int i = tid; i < DIM; i += SIM_THREADS) {
            if (__popc(i & (i >> 1)) & 1) {
                float2 a = st[i];
                st[i] = make_float2(-a.x, -a.y);
            }
        }
        __syncthreads();
    }

    // <Z_q> = sum_idx |amp|^2 * (bit(idx,14-q) ? -1 : +1)
    float ev[N_QUBITS];
#pragma unroll
    for (int q = 0; q < N_QUBITS; ++q) ev[q] = 0.0f;

    for (int i = tid; i < DIM; i += SIM_THREADS) {
        float2 a = st[i];
        float p = a.x * a.x + a.y * a.y;
#pragma unroll
        for (int q = 0; q < N_QUBITS; ++q)
            ev[q] += ((i >> (14 - q)) & 1) ? -p : p;
    }

    // wave32 tree reduction
#pragma unroll
    for (int off = 16; off > 0; off >>= 1) {
#pragma unroll
        for (int q = 0; q < N_QUBITS; ++q)
            ev[q] += __shfl_down(ev[q], off, 32);
    }
    if ((tid & 31) == 0) {
#pragma unroll
        for (int q = 0; q < N_QUBITS; ++q)
            atomicAdd(&acc[q], ev[q]);           // ds_add_f32
    }
    __syncthreads();

    if (tid < N_QUBITS)
        ev_out[tid] = acc[tid] * scale[0];
}

// ---------------------------------------------------------------------------
// Kernel 2: broadcast [15] -> [batch,15] as a rank-1 GEMM on the WMMA pipe.
//   D(16x16) = A(16x4) x B(4x16), A = ones in column K=0, B row K=0 = expvals.
//   One wave emits one 16-row tile via V_WMMA_F32_16X16X4_F32.
// A layout (ISA 32-bit 16x4): lanes 0-15 {VGPR0=K0, VGPR1=K1},
//                             lanes 16-31 {VGPR0=K2, VGPR1=K3}.
// B layout (4x16 row-striped): lanes 0-15 {VGPR0=K0 row}, lanes 16-31 {K2,K3}.
// D layout: VGPR r: lanes 0-15 -> (M=r, N=lane), lanes 16-31 -> (M=r+8, N=lane-16).
// ---------------------------------------------------------------------------
__global__ void __launch_bounds__(256)
qbcast_wmma_kernel(const float* __restrict__ ev,  // [15] scaled expvals
                   float* __restrict__ out,       // [batch, 15]
                   int batch)
{
    const int lane  = threadIdx.x & 31;
    const int waveG = (blockIdx.x * blockDim.x + threadIdx.x) >> 5;

    float bval = (lane < N_QUBITS) ? ev[lane] : 0.0f;

    v2f a;
    a[0] = (lane < 16) ? 1.0f : 0.0f;   // A[:,K=0] = 1, rest 0
    a[1] = 0.0f;
    v2f b;
    b[0] = (lane < 16) ? bval : 0.0f;   // B[K=0, n] = ev[n]
    b[1] = 0.0f;
    v8f c = {};

    // EXEC is all-1s here (no divergence at this point) as WMMA requires.
    v8f d = __builtin_amdgcn_wmma_f32_16x16x4_f32(
        /*neg_a=*/false, a, /*neg_b=*/false, b,
        /*c_mod=*/(short)0, c, /*reuse_a=*/false, /*reuse_b=*/false);

    const int col  = lane & 15;
    const int row0 = waveG * 16 + ((lane < 16) ? 0 : 8);
    if (col < N_QUBITS) {
#pragma unroll
        for (int r = 0; r < 8; ++r) {
            int row = row0 + r;
            if (row < batch)
                out[row * N_QUBITS + col] = d[r];
        }
    }
}

extern "C" void kernel_launch(void* const* d_in, const int* in_sizes, int n_in,
                              void* d_out, int out_size, void* d_ws, size_t ws_size,
                              hipStream_t stream) {
    (void)in_sizes; (void)n_in; (void)ws_size;

    const float* weights = (const float*)d_in[0];  // [2,15,3] f32
    const float* scale   = (const float*)d_in[1];  // [1] f32
    float* out = (float*)d_out;                    // [batch, 15] f32
    float* ev  = (float*)d_ws;                     // 15-float scratch

    const int batch = out_size / N_QUBITS;         // 8192

    // Phase 1: whole circuit in one WGP's LDS (256 KB state + 64 B reduce).
    const size_t lds_bytes = (size_t)(2 * DIM + 16) * sizeof(float);
    qsim_kernel<<<1, SIM_THREADS, lds_bytes, stream>>>(weights, scale, ev);

    // Phase 2: WMMA rank-1 broadcast, one 16-row tile per wave.
    const int nwaves   = (batch + 15) / 16;        // 512
    const int nthreads = nwaves * 32;              // 16384
    const int nblocks  = (nthreads + 255) / 256;   // 64
    qbcast_wmma_kernel<<<nblocks, 256, 0, stream>>>(ev, out, batch);
}